// HingeAP_73650099192604
// MI455X (gfx1250) — compile-verified
//
#include <hip/hip_runtime.h>

typedef __attribute__((ext_vector_type(2))) float v2f;
typedef __attribute__((ext_vector_type(8))) float v8f;

#define BMAX 2048
#define NEG_INF_F (-1e30f)

// One block per class. B <= BMAX assumed (reference: B=2048, C=16).
__global__ __launch_bounds__(256)
void ap_hinge_class_kernel(const float* __restrict__ pred,
                           const int* __restrict__ targ,
                           const int* __restrict__ mask,
                           float* __restrict__ per_class,
                           int B, int C) {
    __shared__ float pos_s[BMAX];   // positives (desc sorted), -inf padded
    __shared__ float neg_s[BMAX];   // negatives (desc sorted), -inf padded
    __shared__ int   sP, sN;
    __shared__ float wave_part[8];

    const int tid  = threadIdx.x;
    const int c    = blockIdx.x;
    const int lane = tid & 31;
    const int wave = tid >> 5;

    if (tid == 0) { sP = 0; sN = 0; }
    __syncthreads();

    // ---- load class column, split into pos/neg arrays, count P/N ----
    int cp = 0, cn = 0;
    for (int i = tid; i < BMAX; i += 256) {
        float pv = NEG_INF_F, nv = NEG_INF_F;
        if (i < B) {
            float s = pred[i * C + c];
            int   t = targ[i * C + c];
            int   m = mask[i * C + c];
            bool p = (t == 1) && (m != 0);
            bool n = (t == 0) && (m != 0);
            pv = p ? s : NEG_INF_F;
            nv = n ? s : NEG_INF_F;
            cp += p ? 1 : 0;
            cn += n ? 1 : 0;
        }
        pos_s[i] = pv;
        neg_s[i] = nv;
    }
    for (int off = 16; off > 0; off >>= 1) {
        cp += __shfl_down(cp, off, 32);
        cn += __shfl_down(cn, off, 32);
    }
    if (lane == 0) { atomicAdd(&sP, cp); atomicAdd(&sN, cn); }
    __syncthreads();

    const int P = sP;
    const int N = sN;

    // ---- bitonic sort both arrays descending (in LDS) ----
    for (unsigned k = 2u; k <= (unsigned)BMAX; k <<= 1) {
        for (unsigned st = k >> 1; st > 0u; st >>= 1) {
            __syncthreads();
            for (unsigned i = (unsigned)tid; i < (unsigned)BMAX; i += 256u) {
                unsigned j = i ^ st;
                if (j > i) {
                    bool upper = ((i & k) == 0u);   // descending half
                    float ai = pos_s[i], aj = pos_s[j];
                    if (upper ? (ai < aj) : (aj < ai)) { pos_s[i] = aj; pos_s[j] = ai; }
                    float bi = neg_s[i], bj = neg_s[j];
                    if (upper ? (bi < bj) : (bj < bi)) { neg_s[i] = bj; neg_s[j] = bi; }
                }
            }
        }
    }
    __syncthreads();

    const float Ps    = fmaxf((float)P, 1.0f);
    const float Ns    = fmaxf((float)N, 1.0f);
    const float invPs = 1.0f / Ps;
    const float c2    = 2.0f / (Ps * Ns);

    const int half = lane >> 4;     // row-half of the C/D tile this lane holds
    const int jl   = lane & 15;     // column within tile

    const int KT = (P + 15) >> 4;   // k tiles (suffix direction)
    const int NT = (N + 15) >> 4;   // column tiles

    float wave_loss = 0.0f;

    // Each wave owns column tiles jt, jt+8, ... (uniform across wave -> EXEC full)
    for (int jt = wave; jt < NT; jt += 8) {
        const int   j0     = jt * 16 + jl;              // 0-based negative rank
        const float jvalf  = (j0 < N) ? 1.0f : 0.0f;    // column-valid flag
        const float nj     = neg_s[j0];                 // in-bounds by construction
        const float jf     = (float)(j0 + 1);
        const float c2nj   = c2 * nj;                   // folded per-column constant

        float carry = 0.0f;   // suffix sum of b below current k tile (per column)
        float cmax  = 0.0f;   // running max(S) clamped at 0 (matches h = max(.,0))

        for (int kt = KT - 1; kt >= 0; --kt) {
            const int k0 = kt * 16;

            v8f acc;
#pragma unroll
            for (int r = 0; r < 8; ++r) acc[r] = carry;

            // S_tile = T(16x16 upper-tri ones) * b_tile + carry, via 4x K=4 WMMA
#pragma unroll
            for (int cc = 0; cc < 4; ++cc) {
                // per documented f32 layouts: K index = vgpr + 2*half
                const int kk = cc * 4 + half * 2;       // local K for element .x
                v2f A, Bv;
                A.x = (kk     >= jl) ? 1.0f : 0.0f;     // T(row=jl, col=kk)
                A.y = (kk + 1 >= jl) ? 1.0f : 0.0f;
#pragma unroll
                for (int e = 0; e < 2; ++e) {
                    const int   kr = k0 + kk + e;       // global 0-based positive rank
                    // branchless b(k,j):
                    //   delta = k/(k+j-1) - k/(k+j) = k / ((k+j-1)(k+j))
                    const float kf  = (float)(kr + 1);
                    const float sm  = kf + jf;
                    const float dl  = kf * invPs *
                                      __builtin_amdgcn_rcpf((sm - 1.0f) * sm);
                    const float ps  = pos_s[kr];        // in-bounds by construction
                    const float raw = dl - c2 * ps + c2nj;
                    const float bv  = (kr < P) ? (raw * jvalf) : 0.0f;
                    if (e == 0) Bv.x = bv; else Bv.y = bv;
                }
                acc = __builtin_amdgcn_wmma_f32_16x16x4_f32(
                    false, A, false, Bv, (short)0, acc, false, false);
            }

            float rmax = acc[0];
#pragma unroll
            for (int r = 1; r < 8; ++r) rmax = fmaxf(rmax, acc[r]);
            cmax = fmaxf(cmax, rmax);

            // new carry = S(k0, j) = full tile sum + old carry; lives in half-0 acc[0]
            carry = __shfl(acc[0], jl, 32);
        }

        // combine the two row-halves of the column; clamp folded into cmax init
        cmax = fmaxf(cmax, __shfl_xor(cmax, 16, 32));
        wave_loss += (half == 0) ? (cmax * jvalf) : 0.0f;
    }

    // deterministic reduction: shuffle tree, then per-wave partials in LDS
    for (int off = 16; off > 0; off >>= 1)
        wave_loss += __shfl_down(wave_loss, off, 32);
    if (lane == 0) wave_part[wave] = wave_loss;
    __syncthreads();

    if (tid == 0) {
        float L = 0.0f;
        for (int w = 0; w < 8; ++w) L += wave_part[w];
        per_class[c] = (P > 0 && N > 0) ? L : 0.0f;
    }
}

__global__ void ap_hinge_mean_kernel(const float* __restrict__ per_class,
                                     float* __restrict__ out, int C) {
    if (threadIdx.x == 0 && blockIdx.x == 0) {
        float s = 0.0f;
        for (int i = 0; i < C; ++i) s += per_class[i];
        out[0] = s / (float)C;
    }
}

extern "C" void kernel_launch(void* const* d_in, const int* in_sizes, int n_in,
                              void* d_out, int out_size, void* d_ws, size_t ws_size,
                              hipStream_t stream) {
    const float* pred = (const float*)d_in[0];
    const int*   targ = (const int*)d_in[1];
    const int*   mask = (const int*)d_in[2];

    const int C = 16;
    const int B = in_sizes[0] / C;   // 2048

    float* per_class = (float*)d_ws;   // 16 floats of scratch

    ap_hinge_class_kernel<<<C, 256, 0, stream>>>(pred, targ, mask, per_class, B, C);
    ap_hinge_mean_kernel<<<1, 32, 0, stream>>>(per_class, (float*)d_out, C);
}